// PoolingAggregator_5076651344591
// MI455X (gfx1250) — compile-verified
//
#include <hip/hip_runtime.h>

typedef __attribute__((ext_vector_type(2))) float v2f;
typedef __attribute__((ext_vector_type(8))) float v8f;

#define DIM 128      // feature dim D, also output dim O
#define KN  32       // neighbours per node

// ---------------------------------------------------------------------------
// Kernel 1: per node, h = relu(neigh_feat @ W1 + b1) ; pooled = mean_k h
// One wave per node. fp32 WMMA 16x16x4, K-dim = 128 -> 32 k-steps.
// W1 staged in LDS with k-pair interleaving: entry (p, c) = float2
// { W1[2p][c], W1[2p+1][c] }  ->  every B fragment is ONE ds_load_b64.
// ---------------------------------------------------------------------------
__global__ __launch_bounds__(128) void pool_wmma_kernel(
    const float* __restrict__ features,
    const float* __restrict__ W1,
    const float* __restrict__ b1,
    const int*  __restrict__ neighbours,
    float* __restrict__ pooled,
    int nnodes)
{
    __shared__ float sW1[DIM * DIM];           // 64 KB, pair-interleaved
    const int tid = threadIdx.x;

    // stage W1 pair-interleaved: sW1[p*256 + c*2 + j] = W1[2p+j][c]
    // Each iteration handles rows (2p, 2p+1), 4 columns -> 2 b128 loads,
    // 2 b128 LDS stores.
    for (int i = tid; i < (DIM / 2) * (DIM / 4); i += 128) {
        const int p  = i >> 5;                 // k-pair index 0..63
        const int c4 = (i & 31) << 2;          // column 0..124 step 4
        const float4 a = *(const float4*)(W1 + (2 * p)     * DIM + c4);
        const float4 b = *(const float4*)(W1 + (2 * p + 1) * DIM + c4);
        float4 q0 = {a.x, b.x, a.y, b.y};
        float4 q1 = {a.z, b.z, a.w, b.w};
        *(float4*)(sW1 + p * 2 * DIM + c4 * 2)     = q0;
        *(float4*)(sW1 + p * 2 * DIM + c4 * 2 + 4) = q1;
    }
    __syncthreads();

    const int lane = tid & 31;
    const int wave = tid >> 5;
    const int lm   = lane & 15;   // M index within tile / N index within tile
    const int hi   = lane >> 4;   // lane half selects K=0/1 vs K=2/3
    const int koff = hi * 2;

    // per-lane LDS base for B fragments: pair row (ks*2 + hi), column nc
    const float* sB = sW1 + hi * 2 * DIM + lm * 2;

    for (int ni = 0; ni < 4; ++ni) {
        const int node = (blockIdx.x * 4 + wave) * 4 + ni;
        if (node >= nnodes) break;             // wave-uniform

        // this lane's two neighbor rows (M-tile 0: rows 0..15, tile 1: 16..31)
        const int r0 = neighbours[node * KN + lm];
        const int r1 = neighbours[node * KN + 16 + lm];
        const float* p0 = features + (size_t)r0 * DIM + koff;
        const float* p1 = features + (size_t)r1 * DIM + koff;

        for (int np = 0; np < 2; ++np) {       // N-tile passes (4 tiles each)
            v8f acc[2][4] = {};
            for (int ks = 0; ks < 32; ++ks) {
                // A fragments: VGPR0 = K(0|2), VGPR1 = K(1|3) per lane half
                v2f a0 = *(const v2f*)(p0 + ks * 4);
                v2f a1 = *(const v2f*)(p1 + ks * 4);
                const float* sBk = sB + ks * 4 * DIM;  // pair rows ks*2 + hi
                for (int n = 0; n < 4; ++n) {
                    // pair (kb, kb+1) at column (np*4+n)*16+lm: one b64 load
                    v2f b = *(const v2f*)(sBk + (np * 4 + n) * 32);
                    acc[0][n] = __builtin_amdgcn_wmma_f32_16x16x4_f32(
                        false, a0, false, b, (short)0, acc[0][n], false, false);
                    acc[1][n] = __builtin_amdgcn_wmma_f32_16x16x4_f32(
                        false, a1, false, b, (short)0, acc[1][n], false, false);
                }
            }
            // fused bias + relu + mean over the 32 neighbor rows
            for (int n = 0; n < 4; ++n) {
                const int nc = (np * 4 + n) * 16 + lm;
                const float bias = b1[nc];
                float s = 0.0f;
                for (int r = 0; r < 8; ++r) {
                    s += fmaxf(acc[0][n][r] + bias, 0.0f);  // rows r / r+8
                    s += fmaxf(acc[1][n][r] + bias, 0.0f);  // rows 16+r / 24+r
                }
                s += __shfl_xor(s, 16);                     // fold lane halves
                if (hi == 0) pooled[(size_t)node * DIM + nc] = s * (1.0f / 32.0f);
            }
        }
    }
}

// ---------------------------------------------------------------------------
// Kernel 2: out = relu(concat[node_feat, pooled] @ Wn), K-dim = 256.
// One wave per 16-row M tile; Wn (256x128, 128 KB) staged in two 64 KB
// column-half passes, pair-interleaved like kernel 1.
// ---------------------------------------------------------------------------
__global__ __launch_bounds__(128) void out_wmma_kernel(
    const float* __restrict__ features,
    const float* __restrict__ pooled,
    const float* __restrict__ Wn,
    const int*  __restrict__ node,
    float* __restrict__ out,
    int nrows)
{
    __shared__ float sWn[256 * 64];            // 64 KB: one column half
    const int tid  = threadIdx.x;
    const int lane = tid & 31;
    const int wave = tid >> 5;
    const int lm   = lane & 15;
    const int hi   = lane >> 4;
    const int koff = hi * 2;

    const int m0 = (blockIdx.x * 4 + wave) * 16;
    const bool active = (m0 + 16 <= nrows);    // wave-uniform

    const float* pA = features;
    const float* pP = pooled;
    if (active) {
        const int row  = m0 + lm;
        const int nidx = node[row];
        pA = features + (size_t)nidx * DIM + koff;
        pP = pooled   + (size_t)row  * DIM + koff;
    }

    for (int np = 0; np < 2; ++np) {
        __syncthreads();
        // stage Wn rows 0..255, columns [np*64, np*64+64) pair-interleaved:
        // sWn[p*128 + c*2 + j] = Wn[2p+j][np*64 + c],  p = 0..127, c = 0..63
        for (int i = tid; i < 128 * 16; i += 128) {
            const int p  = i >> 4;             // k-pair 0..127
            const int c4 = (i & 15) << 2;      // column 0..60 step 4
            const float4 a = *(const float4*)(Wn + (2 * p)     * DIM + np * 64 + c4);
            const float4 b = *(const float4*)(Wn + (2 * p + 1) * DIM + np * 64 + c4);
            float4 q0 = {a.x, b.x, a.y, b.y};
            float4 q1 = {a.z, b.z, a.w, b.w};
            *(float4*)(sWn + p * 128 + c4 * 2)     = q0;
            *(float4*)(sWn + p * 128 + c4 * 2 + 4) = q1;
        }
        __syncthreads();

        if (active) {
            const float* sB = sWn + hi * 128 + lm * 2;
            v8f acc[4] = {};
            for (int ks = 0; ks < 64; ++ks) {
                v2f a;
                if (ks < 32) a = *(const v2f*)(pA + ks * 4);
                else         a = *(const v2f*)(pP + (ks - 32) * 4);
                const float* sBk = sB + ks * 256;   // pair rows ks*2 + hi
                for (int n = 0; n < 4; ++n) {
                    v2f b = *(const v2f*)(sBk + n * 32);
                    acc[n] = __builtin_amdgcn_wmma_f32_16x16x4_f32(
                        false, a, false, b, (short)0, acc[n], false, false);
                }
            }
            for (int n = 0; n < 4; ++n) {
                const int nc = np * 64 + n * 16 + lm;
                for (int r = 0; r < 8; ++r) {
                    const int M = r + hi * 8;  // C layout: VGPR r = row r / r+8
                    out[(size_t)(m0 + M) * DIM + nc] = fmaxf(acc[n][r], 0.0f);
                }
            }
        }
    }
}

// ---------------------------------------------------------------------------
// Kernel 3: raw_features pass-through (second tuple output)
// ---------------------------------------------------------------------------
__global__ void copy4_kernel(const float4* __restrict__ src,
                             float4* __restrict__ dst, int n4)
{
    const int i = blockIdx.x * blockDim.x + threadIdx.x;
    if (i < n4) dst[i] = src[i];
}

extern "C" void kernel_launch(void* const* d_in, const int* in_sizes, int n_in,
                              void* d_out, int out_size, void* d_ws, size_t ws_size,
                              hipStream_t stream) {
    const float* features     = (const float*)d_in[0];
    const float* raw_features = (const float*)d_in[1];
    const float* W1           = (const float*)d_in[2];
    const float* b1           = (const float*)d_in[3];
    const float* Wn           = (const float*)d_in[4];
    const int*   node         = (const int*)d_in[5];
    const int*   neighbours   = (const int*)d_in[6];

    const int B    = in_sizes[5];          // 20000 nodes
    const int rawN = in_sizes[1];          // N*D floats to pass through

    float* out    = (float*)d_out;         // (B, 128) relu output first
    float* rawOut = out + (size_t)B * DIM; // then raw_features copy
    float* pooled = (float*)d_ws;          // (B, 128) scratch

    // Kernel 1: 16 nodes per block (4 waves x 4 nodes)
    const int blk1 = (B + 15) / 16;
    pool_wmma_kernel<<<blk1, 128, 0, stream>>>(features, W1, b1, neighbours,
                                               pooled, B);

    // Kernel 2: 4 M-tiles (64 rows) per block
    const int mtiles = (B + 15) / 16;
    const int blk2   = (mtiles + 3) / 4;
    out_wmma_kernel<<<blk2, 128, 0, stream>>>(features, pooled, Wn, node,
                                              out, B);

    // Kernel 3: raw passthrough
    const int n4   = rawN / 4;
    const int blk3 = (n4 + 255) / 256;
    copy4_kernel<<<blk3, 256, 0, stream>>>((const float4*)raw_features,
                                           (float4*)rawOut, n4);
}